// VectorQuantizer_57578331570266
// MI455X (gfx1250) — compile-verified
//
#include <hip/hip_runtime.h>

#define K_CODES 4096
#define CDIM 128
#define ROWS_PB 128       // pixels (M rows) per block: 8 waves x 16 rows
#define NCHUNK 64         // codebook columns staged per chunk
#define NCHUNKS (K_CODES / NCHUNK)
#define LDSPAD 132        // float stride: conflict-free b64 frag loads

typedef __attribute__((ext_vector_type(2))) float v2f;
typedef __attribute__((ext_vector_type(4))) float v4f;
typedef __attribute__((ext_vector_type(8))) float v8f;

// ---------------- pass 0: e2[k] = ||emb[k]||^2 ----------------
__global__ __launch_bounds__(256) void vq_e2_kernel(const float* __restrict__ emb,
                                                    float* __restrict__ e2g) {
    int k = blockIdx.x * 256 + threadIdx.x;
    const float* row = emb + (size_t)k * CDIM;
    float s = 0.f;
#pragma unroll 8
    for (int c = 0; c < CDIM; ++c) { float v = row[c]; s += v * v; }
    e2g[k] = s;
}

// ---------------- pass 1: fused GEMM + argmin + gather ----------------
__global__ __launch_bounds__(256) void vq_main_kernel(const float* __restrict__ in,    // [32,128,32,32]
                                                      const float* __restrict__ emb,   // [4096,128]
                                                      const float* __restrict__ e2g,   // [4096]
                                                      float* __restrict__ outq,        // [32,128,32,32]
                                                      float* __restrict__ partials) {  // [256]
    __shared__ float b_s[NCHUNK * LDSPAD];       // 33.8 KB codebook chunk (codes x C)
    __shared__ float x2s[ROWS_PB];
    __shared__ int   idx_s[ROWS_PB];
    __shared__ float dmin_s[ROWS_PB];

    const int tid  = threadIdx.x;
    const int lane = tid & 31;
    const int hi   = lane >> 4;      // selects K pair within WMMA frag
    const int l16  = lane & 15;
    const int wave = tid >> 5;

    const int nbase = blockIdx.x * ROWS_PB;            // flat pixel base (within one image)
    const int b     = nbase >> 10;                     // /(H*W)
    const int hw0   = nbase & 1023;
    const size_t img_off = (size_t)b * (CDIM * 1024) + hw0;

    // ---- A fragments straight into registers (reused for all 64 chunks) ----
    // lane(l16,hi) holds row rloc = wave*16+l16, columns {4j+2hi, 4j+2hi+1}
    const int rloc = wave * 16 + l16;
    const float* arow = in + img_off + rloc;
    v2f areg[32];
#pragma unroll
    for (int j = 0; j < 32; ++j) {
        int c0 = 4 * j + 2 * hi;
        areg[j].x = arow[(size_t)c0 * 1024];
        areg[j].y = arow[(size_t)(c0 + 1) * 1024];
    }
    // ||x||^2 : lane and lane^16 jointly hold the full row
    float p = 0.f;
#pragma unroll
    for (int j = 0; j < 32; ++j) p += areg[j].x * areg[j].x + areg[j].y * areg[j].y;
    p += __shfl_xor(p, 16, 32);
    if (hi == 0) x2s[rloc] = p;

    // ---- software-pipelined codebook staging: global -> regs -> LDS ----
    v4f sreg[8];
    auto loadRegs = [&](int chunk) {
        const float* eb = emb + (size_t)chunk * NCHUNK * CDIM;
#pragma unroll
        for (int it = 0; it < 8; ++it) {
            int e4 = it * 256 + tid;                 // vec4 index
            int j  = e4 >> 5;                        // code row in chunk
            int c  = (e4 & 31) << 2;                 // column (x4 floats)
            sreg[it] = *(const v4f*)(eb + (size_t)j * CDIM + c);
        }
    };
    auto storeRegs = [&]() {
#pragma unroll
        for (int it = 0; it < 8; ++it) {
            int e4 = it * 256 + tid;
            int j  = e4 >> 5;
            int c  = (e4 & 31) << 2;
            *(v4f*)(&b_s[j * LDSPAD + c]) = sreg[it];
        }
    };

    loadRegs(0);
    storeRegs();
    __syncthreads();                                  // x2s + first slab visible

    float x2r[8];
#pragma unroll
    for (int v = 0; v < 8; ++v) x2r[v] = x2s[wave * 16 + v + 8 * hi];

    float best[8]; int bestI[8];
#pragma unroll
    for (int v = 0; v < 8; ++v) { best[v] = 3.4e38f; bestI[v] = 0; }

    const float* bfrag = &b_s[l16 * LDSPAD];

    for (int chunk = 0; chunk < NCHUNKS; ++chunk) {
        if (chunk + 1 < NCHUNKS) loadRegs(chunk + 1);     // in flight during WMMAs

        v8f acc0 = {0.f,0.f,0.f,0.f,0.f,0.f,0.f,0.f};
        v8f acc1 = acc0, acc2 = acc0, acc3 = acc0;
#pragma unroll
        for (int j = 0; j < 32; ++j) {                    // K = 128, 4 per WMMA
            int off = 4 * j + 2 * hi;
            v2f b0 = *(const v2f*)(bfrag + off);
            v2f b1 = *(const v2f*)(bfrag + 16 * LDSPAD + off);
            v2f b2 = *(const v2f*)(bfrag + 32 * LDSPAD + off);
            v2f b3 = *(const v2f*)(bfrag + 48 * LDSPAD + off);
            acc0 = __builtin_amdgcn_wmma_f32_16x16x4_f32(false, areg[j], false, b0, (short)0, acc0, false, false);
            acc1 = __builtin_amdgcn_wmma_f32_16x16x4_f32(false, areg[j], false, b1, (short)0, acc1, false, false);
            acc2 = __builtin_amdgcn_wmma_f32_16x16x4_f32(false, areg[j], false, b2, (short)0, acc2, false, false);
            acc3 = __builtin_amdgcn_wmma_f32_16x16x4_f32(false, areg[j], false, b3, (short)0, acc3, false, false);
        }

        // distances + running argmin (ascending index order => strict '<' keeps lowest idx)
        int kbase = chunk * NCHUNK + l16;
        float e20 = e2g[kbase];
        float e21 = e2g[kbase + 16];
        float e22 = e2g[kbase + 32];
        float e23 = e2g[kbase + 48];
#pragma unroll
        for (int v = 0; v < 8; ++v) {
            float d0 = x2r[v] - 2.0f * acc0[v] + e20;
            if (d0 < best[v]) { best[v] = d0; bestI[v] = kbase; }
            float d1 = x2r[v] - 2.0f * acc1[v] + e21;
            if (d1 < best[v]) { best[v] = d1; bestI[v] = kbase + 16; }
            float d2 = x2r[v] - 2.0f * acc2[v] + e22;
            if (d2 < best[v]) { best[v] = d2; bestI[v] = kbase + 32; }
            float d3 = x2r[v] - 2.0f * acc3[v] + e23;
            if (d3 < best[v]) { best[v] = d3; bestI[v] = kbase + 48; }
        }

        __syncthreads();                                  // all waves done reading b_s
        if (chunk + 1 < NCHUNKS) {
            storeRegs();                                  // ds stores (waits loads internally)
            __syncthreads();                              // new slab visible
        }
    }

    // argmin across the 16 lanes of each half (tile columns), jnp tie-break
#pragma unroll
    for (int v = 0; v < 8; ++v) {
        float d = best[v]; int i = bestI[v];
#pragma unroll
        for (int off = 8; off >= 1; off >>= 1) {
            float od = __shfl_xor(d, off, 32);
            int   oi = __shfl_xor(i, off, 32);
            if (od < d || (od == d && oi < i)) { d = od; i = oi; }
        }
        best[v] = d; bestI[v] = i;
    }
    if (l16 == 0) {
#pragma unroll
        for (int v = 0; v < 8; ++v) {
            int rl = wave * 16 + v + 8 * hi;
            idx_s[rl]  = bestI[v];
            dmin_s[rl] = best[v];
        }
    }
    __syncthreads();

    if (tid == 0) {                                       // deterministic loss partial
        float s = 0.f;
        for (int r = 0; r < ROWS_PB; ++r) s += dmin_s[r];
        partials[blockIdx.x] = s;
    }

    // quantized output in NCHW: out[b, c, hw0+r] = emb[idx[r]][c] (coalesced over r)
#pragma unroll 4
    for (int it = 0; it < 64; ++it) {
        int e = it * 256 + tid;
        int c = e >> 7, r = e & 127;
        outq[img_off + (size_t)c * 1024 + r] = emb[(size_t)idx_s[r] * CDIM + c];
    }
}

// ---------------- pass 2: deterministic loss reduction ----------------
__global__ __launch_bounds__(256) void vq_loss_kernel(const float* __restrict__ partials,
                                                      float* __restrict__ out_loss) {
    __shared__ float s[256];
    int t = threadIdx.x;
    s[t] = partials[t];
    __syncthreads();
    for (int off = 128; off >= 1; off >>= 1) {
        if (t < off) s[t] += s[t + off];
        __syncthreads();
    }
    if (t == 0) *out_loss = 1.25f * s[0] * (1.0f / 4194304.0f);
}

extern "C" void kernel_launch(void* const* d_in, const int* in_sizes, int n_in,
                              void* d_out, int out_size, void* d_ws, size_t ws_size,
                              hipStream_t stream) {
    const float* in  = (const float*)d_in[0];   // [32,128,32,32]
    const float* emb = (const float*)d_in[1];   // [4096,128]
    float* outq      = (float*)d_out;           // 4194304 quantized + 1 loss
    float* partials  = (float*)d_ws;            // [256]
    float* e2g       = partials + 256;          // [4096]
    float* loss_out  = outq + (out_size - 1);

    vq_e2_kernel  <<<K_CODES / 256, 256, 0, stream>>>(emb, e2g);
    vq_main_kernel<<<32768 / ROWS_PB, 256, 0, stream>>>(in, emb, e2g, outq, partials);
    vq_loss_kernel<<<1, 256, 0, stream>>>(partials, loss_out);
}